// FusedScaleMaskSoftmax_77850577207442
// MI455X (gfx1250) — compile-verified
//
#include <hip/hip_runtime.h>
#include <math.h>

// softmax(x*S) == softmax base-2 of x*S*log2(e); fold log2(e) into the scale
// so the exp path is a single raw v_exp_f32 per element.
#define SCALE_L2E (0.08838834764831845f * 1.4426950408889634f)
#define SEQ 2048
#define TPB 256
#define EPT 8  // elements per thread: 256*8 = 2048

typedef float v4f __attribute__((ext_vector_type(4)));

// Exact pointer types the async-LDS builtin expects (probe-confirmed round 1/2):
typedef int v4i_vs __attribute__((vector_size(16)));
typedef __attribute__((address_space(1))) v4i_vs* global_v4i_ptr;
typedef __attribute__((address_space(3))) v4i_vs* lds_v4i_ptr;

// ---- CDNA5 async global->LDS copy (ASYNCcnt-tracked) ----
__device__ __forceinline__ void async_load_b128(const float* g, float* l) {
#if __has_builtin(__builtin_amdgcn_global_load_async_to_lds_b128)
    __builtin_amdgcn_global_load_async_to_lds_b128(
        (global_v4i_ptr)(void*)(const_cast<float*>(g)),
        (lds_v4i_ptr)l,
        0, 0);
#else
    unsigned lds_off =
        (unsigned)(unsigned long long)(__attribute__((address_space(3))) float*)l;
    asm volatile("global_load_async_to_lds_b128 %0, %1, off"
                 :: "v"(lds_off), "v"(g) : "memory");
#endif
}

__device__ __forceinline__ void wait_async0() {
#if __has_builtin(__builtin_amdgcn_s_wait_asynccnt)
    __builtin_amdgcn_s_wait_asynccnt(0);
#else
    asm volatile("s_wait_asynccnt 0" ::: "memory");
#endif
}

// ---- ds_swizzle_b32 XOR butterfly (group-of-32 mode): 1 DS op per step,
// no lane-index VALU math (vs HIP __shfl_xor's mbcnt/cmp/cndmask lowering).
template <int XORMASK>
__device__ __forceinline__ float swz_xor(float x) {
    return __int_as_float(__builtin_amdgcn_ds_swizzle(
        __float_as_int(x), (XORMASK << 10) | 0x1F));
}

__device__ __forceinline__ float wave_max32(float x) {
    x = fmaxf(x, swz_xor<16>(x));
    x = fmaxf(x, swz_xor<8>(x));
    x = fmaxf(x, swz_xor<4>(x));
    x = fmaxf(x, swz_xor<2>(x));
    x = fmaxf(x, swz_xor<1>(x));
    return x;
}

__device__ __forceinline__ float wave_sum32(float x) {
    x += swz_xor<16>(x);
    x += swz_xor<8>(x);
    x += swz_xor<4>(x);
    x += swz_xor<2>(x);
    x += swz_xor<1>(x);
    return x;
}

__global__ __launch_bounds__(TPB)
void FusedScaleMaskSoftmax_77850577207442_kernel(const float* __restrict__ x,
                                                 float* __restrict__ out) {
    __shared__ __align__(16) float lrow[SEQ];
    __shared__ __align__(16) float redmax[TPB / 32];
    __shared__ __align__(16) float redsum[TPB / 32];

    const int t   = threadIdx.x;
    const int row = blockIdx.x;            // 65536 rows
    const int i   = row & (SEQ - 1);       // row index within a 2048x2048 matrix
    const int n   = i + 1;                 // number of valid (unmasked) columns
    const size_t base = (size_t)row * SEQ;
    const int c   = t * EPT;               // this thread's chunk start column
    const int rem = n - c;                 // #valid elements in this chunk (may be <=0)

    const float* g = x + base + c;
    float*       l = &lrow[c];

    // Stage this thread's 32B chunk into LDS via the async copy engine.
    // Chunks fully above the diagonal are skipped (EXEC-masked) -> causal
    // read traffic is ~half of the input.
    if (rem > 0) {
        async_load_b128(g, l);
        async_load_b128(g + 4, l + 4);
    }
    wait_async0();  // producer == consumer per lane: no workgroup barrier needed

    // Pull the chunk into registers (single read of the data; both softmax
    // passes run out of registers from here on).
    float v[EPT];
    {
        v4f a = *(const v4f*)l;
        v4f b = *(const v4f*)(l + 4);
        v[0] = a.x; v[1] = a.y; v[2] = a.z; v[3] = a.w;
        v[4] = b.x; v[5] = b.y; v[6] = b.z; v[7] = b.w;
    }

    const float NEG = -__builtin_inff();
    float m = NEG;
#pragma unroll
    for (int k = 0; k < EPT; ++k) {
        v[k] = (k < rem) ? v[k] * SCALE_L2E : NEG;  // scale + causal mask
        m = fmaxf(m, v[k]);
    }

    // ---- row max: wave32 ds_swizzle butterfly + cross-wave LDS reduce ----
    m = wave_max32(m);
    const int wid = t >> 5;
    if ((t & 31) == 0) redmax[wid] = m;
    __syncthreads();
    float bm;
    {
        v4f ra = *(const v4f*)&redmax[0];
        v4f rb = *(const v4f*)&redmax[4];
        bm = fmaxf(fmaxf(fmaxf(ra.x, ra.y), fmaxf(ra.z, ra.w)),
                   fmaxf(fmaxf(rb.x, rb.y), fmaxf(rb.z, rb.w)));
    }

    // ---- exp2 + row sum ----
    float e[EPT];
    float s = 0.0f;
#pragma unroll
    for (int k = 0; k < EPT; ++k) {
        e[k] = __builtin_amdgcn_exp2f(v[k] - bm);  // masked: exp2(-inf) == 0
        s += e[k];
    }
    s = wave_sum32(s);
    if ((t & 31) == 0) redsum[wid] = s;
    __syncthreads();
    float bs;
    {
        v4f ra = *(const v4f*)&redsum[0];
        v4f rb = *(const v4f*)&redsum[4];
        bs = (ra.x + ra.y) + (ra.z + ra.w) + ((rb.x + rb.y) + (rb.z + rb.w));
    }

    const float inv = __builtin_amdgcn_rcpf(bs);

    // ---- normalize + non-temporal B128 stores (512MB stream; keep it out of L2) ----
    v4f r0 = { e[0] * inv, e[1] * inv, e[2] * inv, e[3] * inv };
    v4f r1 = { e[4] * inv, e[5] * inv, e[6] * inv, e[7] * inv };
    v4f* o = (v4f*)(out + base + c);
    __builtin_nontemporal_store(r0, o);
    __builtin_nontemporal_store(r1, o + 1);
}

extern "C" void kernel_launch(void* const* d_in, const int* in_sizes, int n_in,
                              void* d_out, int out_size, void* d_ws, size_t ws_size,
                              hipStream_t stream) {
    const float* x  = (const float*)d_in[0];
    float* out      = (float*)d_out;
    const int rows  = in_sizes[0] / SEQ;   // 2*16*2048 = 65536 rows of length 2048
    dim3 grid(rows), block(TPB);
    FusedScaleMaskSoftmax_77850577207442_kernel<<<grid, block, 0, stream>>>(x, out);
}